// BudgetSampling_45724221833911
// MI455X (gfx1250) — compile-verified
//
#include <hip/hip_runtime.h>
#include <cstdint>

#define NBINS   4096
#define NTILES  (NBINS / 256)   // 16 tiles per array (cnt, sum)
#define WG      256

#if __has_builtin(__builtin_amdgcn_wmma_f32_16x16x4_f32)
#define HAVE_WMMA_F32 1
#else
#define HAVE_WMMA_F32 0
#endif

typedef float v2f __attribute__((ext_vector_type(2)));
typedef float v8f __attribute__((ext_vector_type(8)));

// ---------------- CDNA5 async-copy helpers (ASYNCcnt path) ----------------
__device__ __forceinline__ void async_load_b128(uint32_t lds_byte_addr, const void* g) {
  uint64_t ga = (uint64_t)g;
  asm volatile("global_load_async_to_lds_b128 %0, %1, off"
               :: "v"(lds_byte_addr), "v"(ga) : "memory");
}
__device__ __forceinline__ void wait_async0() { asm volatile("s_wait_asynccnt 0" ::: "memory"); }
__device__ __forceinline__ void wait_async1() { asm volatile("s_wait_asynccnt 1" ::: "memory"); }

// ---------------- zero scratch (cnt + sum bins) ----------------
__global__ __launch_bounds__(WG) void zero_kernel(unsigned* __restrict__ p, int n) {
  int i = blockIdx.x * WG + threadIdx.x;
  if (i < n) p[i] = 0u;
}

// ---------------- pass 1: histogram of pq (LDS-privatized, async-staged) ----------------
__global__ __launch_bounds__(WG) void hist_kernel(const float* __restrict__ pq, int N,
                                                  unsigned* __restrict__ gcnt,
                                                  float* __restrict__ gsum) {
  __shared__ unsigned lcnt[NBINS];
  __shared__ float    lsum[NBINS];
  __shared__ float4   stage[2][WG];

  const int tid = threadIdx.x;
  for (int b = tid; b < NBINS; b += WG) { lcnt[b] = 0u; lsum[b] = 0.f; }
  __syncthreads();

  const int gid    = blockIdx.x * WG + tid;
  const int stride = gridDim.x * WG;
  const int n4     = N >> 2;
  const float4* p4 = (const float4*)pq;

  // LDS byte addresses of this thread's two stage slots (low 32 bits of the
  // generic pointer are the LDS byte offset on gfx1250).
  uint32_t loff[2];
  loff[0] = (uint32_t)(uint64_t)(uintptr_t)&stage[0][tid];
  loff[1] = (uint32_t)(uint64_t)(uintptr_t)&stage[1][tid];

  auto put = [&](float x) {
    int b = (int)(x * (float)NBINS);
    b = (b < 0) ? 0 : ((b > NBINS - 1) ? NBINS - 1 : b);
    atomicAdd(&lcnt[b], 1u);
    atomicAdd(&lsum[b], x);
  };

  int cur = 0;
  int i = gid;
  if (i < n4) async_load_b128(loff[0], p4 + i);
  for (; i < n4; i += stride) {
    int nx = i + stride;
    if (nx < n4) { async_load_b128(loff[cur ^ 1], p4 + nx); wait_async1(); }
    else         { wait_async0(); }
    float4 v = stage[cur][tid];
    put(v.x); put(v.y); put(v.z); put(v.w);
    cur ^= 1;
  }

  // tail (N not multiple of 4): handled by one thread into its block's LDS hist
  if (blockIdx.x == 0 && tid == 0) {
    for (int r = (n4 << 2); r < N; ++r) put(pq[r]);
  }
  __syncthreads();

  for (int b = tid; b < NBINS; b += WG) {
    unsigned c = lcnt[b];
    float    s = lsum[b];
    if (c) atomicAdd(&gcnt[b], c);
    if (s != 0.f) atomicAdd(&gsum[b], s);
  }
}

// ---------------- pass 2: prefix scan via WMMA triangular matmul ----------------
// Each wave scans 256-element tiles: D1 = T(16x16 lower-tri ones) x X (as 16
// segments of 16, K=16 via 4 chained 16x16x4 f32 WMMAs), then D2 adds the
// strictly-lower-triangular cross-segment carries. Cross-tile carries via LDS.
__global__ __launch_bounds__(WG) void scan_kernel(const unsigned* __restrict__ cntU,
                                                  const float* __restrict__ sumF,
                                                  float* __restrict__ prefCnt,
                                                  float* __restrict__ prefSum) {
  __shared__ float s_carry[2 * NTILES];

  const int tid  = threadIdx.x;
  const int wave = tid >> 5;
  const int lane = tid & 31;
  const int n    = lane & 15;            // B/C/D column owned by this lane
  const int m    = lane & 15;            // A row owned by this lane
  const int koff = (lane < 16) ? 0 : 2;  // K sub-position per A/B VGPR layout

  for (int t = wave; t < 2 * NTILES; t += (WG / 32)) {
    const bool isCnt = (t < NTILES);
    const int  base  = (isCnt ? t : t - NTILES) * 256;
    auto ld = [&](int e) -> float {
      return isCnt ? (float)cntU[base + e] : sumF[base + e];
    };
    float* dst = isCnt ? prefCnt : prefSum;

#if HAVE_WMMA_F32
    // Preload B operand values: segment n, K positions 4j+koff(+1)
    float bx[4], by[4];
#pragma unroll
    for (int j = 0; j < 4; ++j) {
      int k0 = 4 * j + koff;
      bx[j] = ld(n * 16 + k0);
      by[j] = ld(n * 16 + k0 + 1);
    }
    // Chain 1: intra-segment inclusive prefix, A = lower-triangular ones
    v8f acc = {0.f, 0.f, 0.f, 0.f, 0.f, 0.f, 0.f, 0.f};
#pragma unroll
    for (int j = 0; j < 4; ++j) {
      int k0 = 4 * j + koff;
      v2f a; a.x = (k0 <= m) ? 1.f : 0.f; a.y = (k0 + 1 <= m) ? 1.f : 0.f;
      v2f b; b.x = bx[j];                 b.y = by[j];
      acc = __builtin_amdgcn_wmma_f32_16x16x4_f32(false, a, false, b,
                                                  (short)0, acc, false, false);
    }
    // Segment totals s[n'] live in row 15 => VGPR7 of lanes 16..31
    float s7 = acc[7];
    // Chain 2: cross-segment exclusive carries, A rows = broadcast s[k],
    // B = strictly-lower-triangular ones.
    v8f acc2 = {0.f, 0.f, 0.f, 0.f, 0.f, 0.f, 0.f, 0.f};
#pragma unroll
    for (int j = 0; j < 4; ++j) {
      int k0 = 4 * j + koff;
      v2f a; a.x = __shfl(s7, 16 + k0);     a.y = __shfl(s7, 16 + k0 + 1);
      v2f b; b.x = (k0 < n) ? 1.f : 0.f;    b.y = (k0 + 1 < n) ? 1.f : 0.f;
      acc2 = __builtin_amdgcn_wmma_f32_16x16x4_f32(false, a, false, b,
                                                   (short)0, acc2, false, false);
    }
    v8f y = acc + acc2;  // tile-local inclusive prefix, element = n*16 + M(r)
#pragma unroll
    for (int r = 0; r < 8; ++r) {
      int mm = r + ((lane < 16) ? 0 : 8);
      dst[base + n * 16 + mm] = y[r];
    }
    if (lane == 31) s_carry[t] = y[7];   // element 255: n=15, m=15
#else
    if (lane == 0) {
      float run = 0.f;
      for (int e = 0; e < 256; ++e) { run += ld(e); dst[base + e] = run; }
      s_carry[t] = run;
    }
#endif
  }
  __syncthreads();

  // Exclusive scan of tile totals (16 tiles per array): serial, trivial size.
  if (tid == 0) {
    float run = 0.f;
    for (int t = 0; t < NTILES; ++t) { float v = s_carry[t]; s_carry[t] = run; run += v; }
  }
  if (tid == 1) {
    float run = 0.f;
    for (int t = NTILES; t < 2 * NTILES; ++t) { float v = s_carry[t]; s_carry[t] = run; run += v; }
  }
  __syncthreads();

  // Apply cross-tile carries.
  for (int idx = tid; idx < 2 * NBINS; idx += WG) {
    const int   t = idx >> 8;                 // global tile id (cnt tiles then sum tiles)
    const int   e = idx & (NBINS - 1);
    float* arr = (idx < NBINS) ? prefCnt : prefSum;
    arr[e] += s_carry[t];
  }
}

// ---------------- pass 3: bisection on sufficient statistics, O(1)/iter ----------------
__global__ void bisect_kernel(const unsigned* __restrict__ cntU,
                              const float* __restrict__ prefCnt,
                              const float* __restrict__ prefSum,
                              const int* __restrict__ n_iter_p,
                              float* __restrict__ c_out, int N) {
  if (threadIdx.x != 0 || blockIdx.x != 0) return;
  const int    iters = *n_iter_p;
  const double Nd    = (double)N;
  const double Stot  = (double)prefSum[NBINS - 1];

  double c_min = 1.0, c_max = 10000.0, c_med = 0.5 * (1.0 + 10000.0);
  bool done = false;
  for (int it = 0; it < iters; ++it) {
    // mean(clip(pq/20 * c, 0, 1)) from the histogram statistics
    double mean;
    double t  = 20.0 / c_med;           // saturation threshold in pq-space
    double tb = t * (double)NBINS;
    if (tb >= (double)NBINS) {
      mean = (c_med / 20.0) * Stot / Nd;          // nothing saturates (the converged regime)
    } else {
      int    b       = (int)tb;
      double f       = tb - (double)b;
      double Psum_b  = (b > 0) ? (double)prefSum[b - 1] : 0.0;
      double PcThru  = (double)prefCnt[b];
      double cb      = (double)cntU[b];
      double below   = f * cb;                    // uniform split of boundary bin
      double avgpq   = ((double)b + 0.5 * f) / (double)NBINS;
      mean = ((c_med / 20.0) * (Psum_b + below * avgpq)
              + (cb - below) + (Nd - PcThru)) / Nd;
    }
    double m  = mean - 0.3;
    bool   hi = (m > 1e-6) && !done;
    bool   lo = (m < -1e-6) && !done;
    done = done || (!hi && !lo);
    if (hi) c_max = c_med;
    if (lo) c_min = c_med;
    if (hi || lo) c_med = 0.5 * (c_min + c_max);
  }
  double c = (c_med > 1.0) ? c_med : 1.0;
  *c_out = (float)c;
}

// ---------------- pass 4: out = clip(pq/20 * c, 0, 1) ----------------
__global__ __launch_bounds__(WG) void finalize_kernel(const float* __restrict__ pq,
                                                      float* __restrict__ out,
                                                      const float* __restrict__ c_p, int N) {
  const float scale = (*c_p) * (1.0f / 20.0f);
  const int gid    = blockIdx.x * WG + threadIdx.x;
  const int stride = gridDim.x * WG;
  const int n4     = N >> 2;
  const float4* p4 = (const float4*)pq;
  float4*       o4 = (float4*)out;
  for (int i = gid; i < n4; i += stride) {
    float4 v = p4[i];
    v.x = __builtin_amdgcn_fmed3f(v.x * scale, 0.f, 1.f);
    v.y = __builtin_amdgcn_fmed3f(v.y * scale, 0.f, 1.f);
    v.z = __builtin_amdgcn_fmed3f(v.z * scale, 0.f, 1.f);
    v.w = __builtin_amdgcn_fmed3f(v.w * scale, 0.f, 1.f);
    o4[i] = v;
  }
  const int tail = N & 3;
  if (gid < tail) {
    int idx = (n4 << 2) + gid;
    out[idx] = __builtin_amdgcn_fmed3f(pq[idx] * scale, 0.f, 1.f);
  }
}

// ---------------- host launcher ----------------
extern "C" void kernel_launch(void* const* d_in, const int* in_sizes, int n_in,
                              void* d_out, int out_size, void* d_ws, size_t ws_size,
                              hipStream_t stream) {
  const float* pq     = (const float*)d_in[0];
  const int*   n_iter = (const int*)d_in[1];
  float*       out    = (float*)d_out;
  const int    N      = in_sizes[0];

  // workspace layout: cntU[NBINS] | sumF[NBINS] | prefCnt[NBINS] | prefSum[NBINS] | c
  unsigned* cntU    = (unsigned*)d_ws;
  float*    sumF    = (float*)d_ws + 1 * NBINS;
  float*    prefCnt = (float*)d_ws + 2 * NBINS;
  float*    prefSum = (float*)d_ws + 3 * NBINS;
  float*    c_ws    = (float*)d_ws + 4 * NBINS;

  zero_kernel<<<(2 * NBINS + WG - 1) / WG, WG, 0, stream>>>((unsigned*)d_ws, 2 * NBINS);

  const int n4 = N >> 2;
  int histBlocks = 1024;
  if (n4 < histBlocks * WG) histBlocks = (n4 + WG - 1) / WG;
  if (histBlocks < 1) histBlocks = 1;
  hist_kernel<<<histBlocks, WG, 0, stream>>>(pq, N, cntU, sumF);

  scan_kernel<<<1, WG, 0, stream>>>(cntU, sumF, prefCnt, prefSum);

  bisect_kernel<<<1, 32, 0, stream>>>(cntU, prefCnt, prefSum, n_iter, c_ws, N);

  int fb = (n4 + WG - 1) / WG;
  if (fb < 1) fb = 1;
  finalize_kernel<<<fb, WG, 0, stream>>>(pq, out, c_ws, N);
}